// FAGCNPropagation_91139206021466
// MI455X (gfx1250) — compile-verified
//
#include <hip/hip_runtime.h>
#include <hip/hip_bf16.h>

#define HIDDEN   128
#define WAVE     32

typedef __attribute__((ext_vector_type(2))) float v2f;
typedef __attribute__((ext_vector_type(8))) float v8f;

// ---------------------------------------------------------------------------
// Phase 1: gates via V_WMMA_F32_16X16X4_F32.
// One wave computes x1/x2 for 16 consecutive nodes.
//   A (16x4 f32): lane m(0..15) holds row M=m; VGPR0/1 = K{0,1} (lanes 0-15)
//                 or K{2,3} (lanes 16-31).
//   B (4x16 f32): column N=0 is w1, N=1 is w2, rest zero. Same K split.
//   D (16x16 f32): VGPR r, lanes 0-15 -> (M=r, N=lane); lanes 16-31 -> (M=r+8).
// B is built branchlessly: per-lane weight pointer select + per-lane 0/1 scale,
// both hoisted out of the K loop so EXEC never changes around the WMMAs.
// ---------------------------------------------------------------------------
__global__ void fagcn_gate_wmma(const float* __restrict__ x,
                                const float* __restrict__ w1,
                                const float* __restrict__ w2,
                                float* __restrict__ g1,
                                float* __restrict__ g2,
                                int n_nodes) {
  const int lane = threadIdx.x & (WAVE - 1);
  const int wave = blockIdx.x * (blockDim.x >> 5) + (threadIdx.x >> 5);
  const int base = wave * 16;                       // 16 nodes per wave
  if (base >= n_nodes) return;

  const int m  = lane & 15;                         // row (A) / column (B,D)
  const int hi = lane >> 4;                         // K-half selector
  const float* __restrict__ row = x + (size_t)(base + m) * HIDDEN;

  // Branchless B setup (hoisted): column 0 reads w1, column 1 reads w2,
  // all other columns multiply to zero.
  const float* __restrict__ wsel = (m == 1) ? w2 : w1;
  const float bscale = (m < 2) ? 1.0f : 0.0f;

  v8f acc = {};
#pragma unroll
  for (int k0 = 0; k0 < HIDDEN; k0 += 4) {
    const int k = k0 + hi * 2;
    v2f a  = *(const v2f*)(row + k);    // A: two consecutive K of this row
    v2f wv = *(const v2f*)(wsel + k);   // weight pair for this K
    v2f b  = { bscale * wv.x, bscale * wv.y };
    acc = __builtin_amdgcn_wmma_f32_16x16x4_f32(
        /*neg_a=*/false, a, /*neg_b=*/false, b,
        /*c_mod=*/(short)0, acc, /*reuse_a=*/false, /*reuse_b=*/false);
  }

  // Column 0 of D = x1 gates, column 1 = x2 gates.
  if (m < 2) {
    float* __restrict__ dst = (m == 0) ? g1 : g2;
    const int mbase = base + hi * 8;
#pragma unroll
    for (int r = 0; r < 8; ++r) dst[mbase + r] = acc[r];
  }
}

// ---------------------------------------------------------------------------
// Zero-fill output (harness poisons d_out with 0xAA before timing).
// ---------------------------------------------------------------------------
__global__ void fagcn_zero(float4* __restrict__ out, size_t n4) {
  size_t i      = (size_t)blockIdx.x * blockDim.x + threadIdx.x;
  size_t stride = (size_t)gridDim.x * blockDim.x;
  const float4 z = {0.f, 0.f, 0.f, 0.f};
  for (; i < n4; i += stride) out[i] = z;
}

// ---------------------------------------------------------------------------
// Phase 2: one wave per edge. Lane holds a float4 of the gathered row
// (coalesced 512B global_load_b128 per edge) and issues 4 global_atomic_add_f32
// into the destination row. Entire working set (~122MB) is L2-resident (192MB).
// ---------------------------------------------------------------------------
__global__ void fagcn_edge_scatter(const float* __restrict__ x,
                                   const float* __restrict__ g1,
                                   const float* __restrict__ g2,
                                   const float* __restrict__ adj,
                                   const int* __restrict__ in_idx,
                                   const int* __restrict__ out_idx,
                                   float* __restrict__ out,
                                   int n_edges) {
  const int lane   = threadIdx.x & (WAVE - 1);
  const int wave   = blockIdx.x * (blockDim.x >> 5) + (threadIdx.x >> 5);
  const int nwaves = gridDim.x * (blockDim.x >> 5);

  for (int e = wave; e < n_edges; e += nwaves) {
    const int en = e + nwaves;
    if (en < n_edges) {                 // warm the next edge records in L2
      __builtin_prefetch(in_idx + en, 0, 0);
      __builtin_prefetch(out_idx + en, 0, 0);
      __builtin_prefetch(adj + en, 0, 0);
    }
    const int src = __builtin_amdgcn_readfirstlane(in_idx[e]);
    const int dst = __builtin_amdgcn_readfirstlane(out_idx[e]);
    const float mval = tanhf(g1[src] + g2[dst]) * adj[e];

    const float4 v = ((const float4*)(x + (size_t)dst * HIDDEN))[lane];
    float* __restrict__ o = out + (size_t)src * HIDDEN + lane * 4;
    unsafeAtomicAdd(o + 0, mval * v.x);
    unsafeAtomicAdd(o + 1, mval * v.y);
    unsafeAtomicAdd(o + 2, mval * v.z);
    unsafeAtomicAdd(o + 3, mval * v.w);
  }
}

// ---------------------------------------------------------------------------
// Host launcher
// Inputs: 0=x [N,128] f32, 1=w1 [128] f32, 2=w2 [128] f32,
//         3=adj_vals [E] f32, 4=in_idx [E] i32, 5=out_idx [E] i32
// Output: [N,128] f32
// ---------------------------------------------------------------------------
extern "C" void kernel_launch(void* const* d_in, const int* in_sizes, int n_in,
                              void* d_out, int out_size, void* d_ws, size_t ws_size,
                              hipStream_t stream) {
  const float* x   = (const float*)d_in[0];
  const float* w1  = (const float*)d_in[1];
  const float* w2  = (const float*)d_in[2];
  const float* adj = (const float*)d_in[3];
  const int* in_i  = (const int*)d_in[4];
  const int* out_i = (const int*)d_in[5];
  float* out       = (float*)d_out;

  const int n_nodes = in_sizes[0] / HIDDEN;
  const int n_edges = in_sizes[3];

  float* g1 = (float*)d_ws;            // [n_nodes]
  float* g2 = g1 + n_nodes;            // [n_nodes]

  // Phase 1: gate GEMV via f32 WMMA (16 nodes per wave, 8 waves per block)
  {
    const int waves  = (n_nodes + 15) / 16;
    const int blocks = (waves + 7) / 8;
    fagcn_gate_wmma<<<blocks, 256, 0, stream>>>(x, w1, w2, g1, g2, n_nodes);
  }

  // Zero output
  {
    const size_t n4 = (size_t)out_size / 4;
    fagcn_zero<<<2048, 256, 0, stream>>>((float4*)out, n4);
  }

  // Phase 2: edge gather + atomic scatter
  {
    fagcn_edge_scatter<<<4096, 256, 0, stream>>>(x, g1, g2, adj, in_i, out_i,
                                                 out, n_edges);
  }
}